// GatNet_64991445123401
// MI455X (gfx1250) — compile-verified
//
#include <hip/hip_runtime.h>
#include <math.h>

typedef __attribute__((ext_vector_type(2))) float v2f;
typedef __attribute__((ext_vector_type(8))) float v8f;

#define FDIM 128
#define NHEAD 8
#define FHID 16
#define NEG_SLOPE 0.2f

// ---------------------------------------------------------------------------
// GEMM: C[M,128] = A[M,128] @ B[128,128], f32 WMMA 16x16x4, one wave per tile.
// Block = 256 threads = 8 waves -> covers one 16-row strip across all 8 col tiles.
// ---------------------------------------------------------------------------
__global__ void gemm_wmma_f32(const float* __restrict__ A,
                              const float* __restrict__ B,
                              float* __restrict__ C) {
    const int K = FDIM, NC = FDIM;
    const int wave = threadIdx.x >> 5;
    const int lane = threadIdx.x & 31;
    const int m0 = blockIdx.x * 16;
    const int n0 = wave * 16;
    const int half = lane >> 4;           // 0: K pair {k,k+1}, 1: {k+2,k+3}
    const int row = m0 + (lane & 15);     // A rows, lanes 0-15 == 16-31
    const int col = n0 + (lane & 15);     // B/C cols

    v8f acc = {};
#pragma unroll 8
    for (int k = 0; k < K; k += 4) {
        // A 16x4 fragment: VGPR0 = K=k (lo lanes) / k+2 (hi), VGPR1 = k+1 / k+3
        v2f a = *(const v2f*)(A + (size_t)row * K + k + 2 * half);
        // B 4x16 fragment, mirrored layout
        v2f b;
        b.x = B[(size_t)(k + 2 * half) * NC + col];
        b.y = B[(size_t)(k + 2 * half + 1) * NC + col];
        acc = __builtin_amdgcn_wmma_f32_16x16x4_f32(
            false, a, false, b, (short)0, acc, false, false);
    }
    // C 16x16: VGPR r -> M = r (lo lanes) / r+8 (hi lanes), N = col
#pragma unroll
    for (int r = 0; r < 8; ++r) {
        C[(size_t)(m0 + r + 8 * half) * NC + col] = acc[r];
    }
}

// ---------------------------------------------------------------------------
// Attention coefficients: al_s[n,h] = <h[n,h,:], a_src[h,:]>, same for al_d.
// One thread per (node, head).
// ---------------------------------------------------------------------------
__global__ void attn_coef(const float* __restrict__ h,
                          const float* __restrict__ a_src,
                          const float* __restrict__ a_dst,
                          float* __restrict__ al_s,
                          float* __restrict__ al_d, int Nn) {
    int idx = blockIdx.x * blockDim.x + threadIdx.x;
    if (idx >= Nn * NHEAD) return;
    int n = idx >> 3, hd = idx & 7;
    const float* hp = h + (size_t)n * FDIM + hd * FHID;
    const float* as = a_src + hd * FHID;
    const float* ad = a_dst + hd * FHID;
    float s = 0.f, d = 0.f;
#pragma unroll
    for (int f = 0; f < FHID; ++f) {
        float v = hp[f];
        s += v * as[f];
        d += v * ad[f];
    }
    al_s[idx] = s;
    al_d[idx] = d;
}

__global__ void zero_f32(float* __restrict__ p, int n) {
    int i = blockIdx.x * blockDim.x + threadIdx.x;
    if (i < n) p[i] = 0.f;
}

// order-preserving float <-> uint mapping (for atomicMax on floats)
__device__ __forceinline__ unsigned f2ord(float f) {
    unsigned u = __float_as_uint(f);
    return (u & 0x80000000u) ? ~u : (u | 0x80000000u);
}
__device__ __forceinline__ float ord2f(unsigned u) {
    return __uint_as_float((u & 0x80000000u) ? (u & 0x7FFFFFFFu) : ~u);
}

__device__ __forceinline__ void edge_endpoints(const long long* __restrict__ ei,
                                               int e, int E_, int& src, int& dst) {
    if (e < E_) {
        src = (int)ei[e];
        dst = (int)ei[(size_t)E_ + e];
    } else {               // appended self-loop
        src = dst = e - E_;
    }
}

// segment max of leaky-relu'd logits over incoming edges of dst (incl self loop)
__global__ void edge_max(const long long* __restrict__ ei,
                         const float* __restrict__ al_s,
                         const float* __restrict__ al_d,
                         unsigned* __restrict__ emax_ord, int E_, int Nn) {
    int e = blockIdx.x * blockDim.x + threadIdx.x;
    if (e >= E_ + Nn) return;
    int src, dst;
    edge_endpoints(ei, e, E_, src, dst);
#pragma unroll
    for (int hd = 0; hd < NHEAD; ++hd) {
        float v = al_s[src * NHEAD + hd] + al_d[dst * NHEAD + hd];
        v = v >= 0.f ? v : NEG_SLOPE * v;
        atomicMax(&emax_ord[dst * NHEAD + hd], f2ord(v));
    }
}

__global__ void ord_to_float(unsigned* __restrict__ p, int n) {
    int i = blockIdx.x * blockDim.x + threadIdx.x;
    if (i < n) {
        unsigned u = p[i];
        ((float*)p)[i] = ord2f(u);
    }
}

// segment sum of exp(e - emax[dst])
__global__ void edge_den(const long long* __restrict__ ei,
                         const float* __restrict__ al_s,
                         const float* __restrict__ al_d,
                         const float* __restrict__ emax,
                         float* __restrict__ den, int E_, int Nn) {
    int e = blockIdx.x * blockDim.x + threadIdx.x;
    if (e >= E_ + Nn) return;
    int src, dst;
    edge_endpoints(ei, e, E_, src, dst);
#pragma unroll
    for (int hd = 0; hd < NHEAD; ++hd) {
        float v = al_s[src * NHEAD + hd] + al_d[dst * NHEAD + hd];
        v = v >= 0.f ? v : NEG_SLOPE * v;
        float w = expf(v - emax[dst * NHEAD + hd]);
        atomicAdd(&den[dst * NHEAD + hd], w);
    }
}

// message pass: one wave per edge; lane l owns features 4l..4l+3 (head = l/4).
// out[dst] += alpha * h[src]   (f32 atomics)
__global__ void edge_msg(const long long* __restrict__ ei,
                         const float* __restrict__ h,
                         const float* __restrict__ al_s,
                         const float* __restrict__ al_d,
                         const float* __restrict__ emax,
                         const float* __restrict__ den,
                         float* __restrict__ out, int E_, int Nn) {
    const int wavesPerBlk = blockDim.x >> 5;
    int e = blockIdx.x * wavesPerBlk + (threadIdx.x >> 5);
    int lane = threadIdx.x & 31;
    if (e >= E_ + Nn) return;
    int src, dst;
    edge_endpoints(ei, e, E_, src, dst);
    int hd = lane >> 2;  // 4 consecutive features never straddle a head
    float v = al_s[src * NHEAD + hd] + al_d[dst * NHEAD + hd];
    v = v >= 0.f ? v : NEG_SLOPE * v;
    float alpha = expf(v - emax[dst * NHEAD + hd]) / den[dst * NHEAD + hd];

    const float4 hv = *(const float4*)(h + (size_t)src * FDIM + lane * 4);
    float* o = out + (size_t)dst * FDIM + lane * 4;
    atomicAdd(o + 0, alpha * hv.x);
    atomicAdd(o + 1, alpha * hv.y);
    atomicAdd(o + 2, alpha * hv.z);
    atomicAdd(o + 3, alpha * hv.w);
}

__global__ void bias_tanh(float* __restrict__ p, const float* __restrict__ b, int n) {
    int i = blockIdx.x * blockDim.x + threadIdx.x;
    if (i < n) p[i] = tanhf(p[i] + b[i & (FDIM - 1)]);
}

__global__ void pool_kernel(const float* __restrict__ h,
                            const long long* __restrict__ batch,
                            float* __restrict__ pooled, int Nn) {
    int i = blockIdx.x * blockDim.x + threadIdx.x;
    if (i >= Nn * FDIM) return;
    int n = i >> 7, c = i & (FDIM - 1);
    int g = (int)batch[n];
    atomicAdd(&pooled[g * FDIM + c], h[i]);
}

__global__ void fc_kernel(const float* __restrict__ pooled,
                          const float* __restrict__ Wfc,
                          const float* __restrict__ bfc,
                          float* __restrict__ out, int nG, int nC) {
    int i = blockIdx.x * blockDim.x + threadIdx.x;
    if (i >= nG * nC) return;
    int g = i / nC, c = i % nC;
    float acc = bfc[c];
    for (int k = 0; k < FDIM; ++k) acc += pooled[g * FDIM + k] * Wfc[k * nC + c];
    out[i] = tanhf(acc);
}

// ---------------------------------------------------------------------------
extern "C" void kernel_launch(void* const* d_in, const int* in_sizes, int n_in,
                              void* d_out, int out_size, void* d_ws, size_t ws_size,
                              hipStream_t stream) {
    const float* x          = (const float*)d_in[0];
    const long long* ei     = (const long long*)d_in[1];
    const long long* batch  = (const long long*)d_in[2];
    const float* W[3]    = {(const float*)d_in[3],  (const float*)d_in[7],  (const float*)d_in[11]};
    const float* Asrc[3] = {(const float*)d_in[4],  (const float*)d_in[8],  (const float*)d_in[12]};
    const float* Adst[3] = {(const float*)d_in[5],  (const float*)d_in[9],  (const float*)d_in[13]};
    const float* Bias[3] = {(const float*)d_in[6],  (const float*)d_in[10], (const float*)d_in[14]};
    const float* Wfc = (const float*)d_in[15];
    const float* bfc = (const float*)d_in[16];

    const int Nn = in_sizes[2];          // 50000
    const int Ee = in_sizes[1] / 2;      // 1600000
    const int ET = Ee + Nn;              // edges + self loops
    const int nC = 10;
    const int nG = out_size / nC;        // 64

    float* bufA   = (float*)d_ws;                    // [N,128] layer input / accumulator
    float* bufB   = bufA + (size_t)Nn * FDIM;        // [N,128] GEMM output h
    float* al_s   = bufB + (size_t)Nn * FDIM;        // [N,8]
    float* al_d   = al_s + (size_t)Nn * NHEAD;       // [N,8]
    float* emax   = al_d + (size_t)Nn * NHEAD;       // [N,8] (uint during max pass)
    float* den    = emax + (size_t)Nn * NHEAD;       // [N,8]
    float* pooled = den  + (size_t)Nn * NHEAD;       // [64,128]

    const int T = 256;
    const float* xin = x;
    for (int layer = 0; layer < 3; ++layer) {
        gemm_wmma_f32<<<Nn / 16, 256, 0, stream>>>(xin, W[layer], bufB);
        attn_coef<<<(Nn * NHEAD + T - 1) / T, T, 0, stream>>>(bufB, Asrc[layer], Adst[layer],
                                                              al_s, al_d, Nn);
        // xin (== bufA for layers 2,3) is dead after the GEMM: reuse as accumulator
        zero_f32<<<(Nn * FDIM + T - 1) / T, T, 0, stream>>>(bufA, Nn * FDIM);
        zero_f32<<<(Nn * NHEAD + T - 1) / T, T, 0, stream>>>(emax, Nn * NHEAD);
        zero_f32<<<(Nn * NHEAD + T - 1) / T, T, 0, stream>>>(den, Nn * NHEAD);
        edge_max<<<(ET + T - 1) / T, T, 0, stream>>>(ei, al_s, al_d, (unsigned*)emax, Ee, Nn);
        ord_to_float<<<(Nn * NHEAD + T - 1) / T, T, 0, stream>>>((unsigned*)emax, Nn * NHEAD);
        edge_den<<<(ET + T - 1) / T, T, 0, stream>>>(ei, al_s, al_d, emax, den, Ee, Nn);
        edge_msg<<<(ET + 7) / 8, T, 0, stream>>>(ei, bufB, al_s, al_d, emax, den, bufA, Ee, Nn);
        bias_tanh<<<(Nn * FDIM + T - 1) / T, T, 0, stream>>>(bufA, Bias[layer], Nn * FDIM);
        xin = bufA;
    }
    zero_f32<<<(nG * FDIM + T - 1) / T, T, 0, stream>>>(pooled, nG * FDIM);
    pool_kernel<<<(Nn * FDIM + T - 1) / T, T, 0, stream>>>(bufA, batch, pooled, Nn);
    fc_kernel<<<(nG * nC + T - 1) / T, T, 0, stream>>>(pooled, Wfc, bfc, (float*)d_out, nG, nC);
}